// fullConv4d_4329327034899
// MI455X (gfx1250) — compile-verified
//
#include <hip/hip_runtime.h>

// CDNA5 / gfx1250 implicit-GEMM 4D convolution using V_WMMA_F32_16X16X4_F32
// with an LDS-staged, zero-padded input slab.
// Shapes: x (2,16,24,24,24,24) f32, w (16,16,3,3,3,3) f32, b (16) f32.
// out (2,16,24,24,24,24) f32, 'same' padding, stride 1.
//
// Block = one (b,h,w): 576 threads = 18 waves, one wave per 32-wide dt tile.
// Per (kh,kw) the block stages x[b][:][hp][wp][:][:] into LDS transposed to
// [pos][cin] with a zero halo border (26x26 padded, cin stride 20 floats for
// bank-conflict-free, 16B-aligned staging). B fragments are then single
// ds_load_b64's from precomputed lane offsets -- no masks needed, the zero
// border supplies the padding. A fragments are packed float2 global loads.
// EXEC is all-ones at every WMMA (no divergence anywhere).

typedef __attribute__((ext_vector_type(2))) float v2f;
typedef __attribute__((ext_vector_type(8))) float v8f;

#define SS    24
#define SS2   576        // 24^2
#define SS3   13824      // 24^3
#define CSTR  331776     // 24^4    channel stride (elements)
#define BSTR  5308416    // 16*24^4 batch stride (elements)
#define NTAP  81
#define CIN   16
#define COUT  16
#define WT_ELEMS (NTAP * CIN * COUT)   // 20736 floats
#define NBLK  1152                      // 2 * 24 * 24  (one block per (b,h,w))
#define TPB   576                       // 18 waves * 32
#define LP    26                        // padded d/t extent (24 + halo)
#define SP    20                        // floats per position in LDS (16 + pad)
#define LDS_FLOATS (LP * LP * SP)       // 13520 floats = 54080 B

// ---------------------------------------------------------------------------
// Weight re-layout: w[co][cin][tap] -> wt[tap][c][khalf][m][j]  (j = 0..1)
//   cin = c*4 + khalf*2 + j.  A lane (khalf = lane>=16, m = lane&15) reads its
//   two A values as one contiguous float2.
// ---------------------------------------------------------------------------
__global__ __launch_bounds__(256) void conv4d_wt_pack(
    const float* __restrict__ w, float* __restrict__ wt)
{
    int i = blockIdx.x * 256 + threadIdx.x;          // 0 .. 20735
    if (i < WT_ELEMS) {
        int j     = i & 1;
        int m     = (i >> 1) & 15;
        int khalf = (i >> 5) & 1;
        int c     = (i >> 6) & 3;
        int tap   = i >> 8;                          // 0 .. 80
        int cin   = c * 4 + khalf * 2 + j;
        wt[i] = w[(m * CIN + cin) * NTAP + tap];
    }
}

// ---------------------------------------------------------------------------
// Main kernel.
// ---------------------------------------------------------------------------
template <bool TR>
__global__ __launch_bounds__(TPB) void conv4d_wmma_f32(
    const float* __restrict__ x,
    const float* __restrict__ wt,     // TR: packed layout; else raw [co][cin][tap]
    const float* __restrict__ bias,
    float* __restrict__ out)
{
    __shared__ float lds[LDS_FLOATS];

    const int tid  = threadIdx.x;
    const int lane = tid & 31;
    const int dtb  = tid >> 5;                       // wave id = dt tile, 0..17

    // block decode: (b, h, w)
    const int bq = blockIdx.x / (SS * SS);
    const int rem = blockIdx.x % (SS * SS);
    const int hq = rem / SS;
    const int wq = rem % SS;

    // per-lane fragment coords (ISA 7.12.2):
    //   A 16x4: row M = lane&15, holds K = 2*hi, 2*hi+1   (hi = lane>=16)
    //   B 4x16: col N = lane&15, holds K = 2*hi, 2*hi+1
    //   C/D:    vgpr r -> (M = r + 8*hi, N = lane&15)
    const int n      = lane & 15;
    const int hi     = lane >> 4;
    const int m      = n;
    const int wlane2 = hi * 32 + m * 2;              // A-pair lane offset (elements)

    // two dt positions per lane (tile0: p0, tile1: p0+16)
    const int p0 = dtb * 32 + n;
    const int p1 = p0 + 16;
    const int d0 = p0 / SS, t0 = p0 % SS;
    const int d1 = p1 / SS, t1 = p1 % SS;

    // per-(kd,kt) LDS element offsets (zero-padded slab => always valid)
    int lofs0[9], lofs1[9];
#pragma unroll
    for (int kd = 0; kd < 3; ++kd) {
#pragma unroll
        for (int kt = 0; kt < 3; ++kt) {
            const int t9 = kd * 3 + kt;
            lofs0[t9] = ((d0 + kd) * LP + (t0 + kt)) * SP + 2 * hi;
            lofs1[t9] = ((d1 + kd) * LP + (t1 + kt)) * SP + 2 * hi;
        }
    }

    // zero the whole LDS slab once (border stays zero; interior is restaged)
    {
        float4* lds4 = (float4*)lds;
        const float4 z = {0.f, 0.f, 0.f, 0.f};
        for (int i = tid; i < LDS_FLOATS / 4; i += TPB) lds4[i] = z;
    }

    // staging coords for this thread (one (d,t) position, all 16 cin)
    const int sd = tid / SS;                         // 0..23
    const int st = tid % SS;
    float* const sdst = &lds[((sd + 1) * LP + (st + 1)) * SP];

    v8f acc0 = {0.f, 0.f, 0.f, 0.f, 0.f, 0.f, 0.f, 0.f};
    v8f acc1 = {0.f, 0.f, 0.f, 0.f, 0.f, 0.f, 0.f, 0.f};

    for (int kh = 0; kh < 3; ++kh) {
        const int hp = hq + kh - 1;
        if ((unsigned)hp >= (unsigned)SS) continue;          // block-uniform skip
        for (int kw = 0; kw < 3; ++kw) {
            const int wp = wq + kw - 1;
            if ((unsigned)wp >= (unsigned)SS) continue;      // block-uniform skip

            // ---- stage slab x[b][:][hp][wp][:][:] -> lds[pos][cin] ----
            __syncthreads();                                 // prior reads done
            {
                const float* xs = x + (bq * BSTR + hp * SS3 + wp * SS2) + tid;
                float4 buf[4];
#pragma unroll
                for (int c4 = 0; c4 < 4; ++c4) {
                    buf[c4].x = xs[(c4 * 4 + 0) * CSTR];
                    buf[c4].y = xs[(c4 * 4 + 1) * CSTR];
                    buf[c4].z = xs[(c4 * 4 + 2) * CSTR];
                    buf[c4].w = xs[(c4 * 4 + 3) * CSTR];
                }
#pragma unroll
                for (int c4 = 0; c4 < 4; ++c4)
                    *(float4*)(sdst + c4 * 4) = buf[c4];     // 16B aligned (SP=20)
            }
            __syncthreads();                                 // slab visible

            // ---- compute: 9 taps x 4 cin-chunks x 2 tiles ----
            const float* wu = TR ? (wt + (kh * 3 + kw) * 9 * 256) : wt;
            const int tapb  = (kh * 3 + kw) * 9;
#pragma unroll
            for (int c = 0; c < 4; ++c) {
#pragma unroll
                for (int t9 = 0; t9 < 9; ++t9) {
                    v2f a;
                    if (TR) {
                        a = *(const v2f*)(wu + (t9 * 4 + c) * 64 + wlane2);
                    } else {
                        const int cin0 = c * 4 + 2 * hi;
                        const int ia   = (m * CIN + cin0) * NTAP + tapb + t9;
                        a.x = wt[ia];
                        a.y = wt[ia + NTAP];
                    }
                    // B fragments: ds_load_b64, zero-padded => no masks
                    const v2f bb0 = *(const v2f*)&lds[lofs0[t9] + c * 4];
                    const v2f bb1 = *(const v2f*)&lds[lofs1[t9] + c * 4];
                    acc0 = __builtin_amdgcn_wmma_f32_16x16x4_f32(
                               false, a, false, bb0, (short)0, acc0, false, false);
                    acc1 = __builtin_amdgcn_wmma_f32_16x16x4_f32(
                               false, a, false, bb1, (short)0, acc1, false, false);
                }
            }
        }
    }

    // write back 16x32 tile + bias
    const int obase = bq * BSTR + hq * SS3 + wq * SS2 + dtb * 32 + n;
#pragma unroll
    for (int r = 0; r < 8; ++r) {
        const int co = r + 8 * hi;
        const float bv = bias[co];
        out[obase + co * CSTR]      = acc0[r] + bv;
        out[obase + co * CSTR + 16] = acc1[r] + bv;
    }
}

// ---------------------------------------------------------------------------
extern "C" void kernel_launch(void* const* d_in, const int* in_sizes, int n_in,
                              void* d_out, int out_size, void* d_ws, size_t ws_size,
                              hipStream_t stream) {
    const float* x    = (const float*)d_in[0];
    const float* w    = (const float*)d_in[1];
    const float* bias = (const float*)d_in[2];
    float* out        = (float*)d_out;

    if (ws_size >= (size_t)WT_ELEMS * sizeof(float)) {
        float* wt = (float*)d_ws;
        conv4d_wt_pack<<<(WT_ELEMS + 255) / 256, 256, 0, stream>>>(w, wt);
        conv4d_wmma_f32<true><<<NBLK, TPB, 0, stream>>>(x, wt, bias, out);
    } else {
        conv4d_wmma_f32<false><<<NBLK, TPB, 0, stream>>>(x, w, bias, out);
    }
}